// Experts_10488310137289
// MI455X (gfx1250) — compile-verified
//
#include <hip/hip_runtime.h>

typedef __attribute__((ext_vector_type(2))) float v2f;
typedef __attribute__((ext_vector_type(8))) float v8f;

#define E_ 32
#define H_ 2048
#define F_ 768
#define T_ 64
#define KB 16    // K slab staged in LDS (double buffered)
#define NB 128   // N tile per block

__device__ __forceinline__ v8f wmma_f32(v2f a, v2f b, v8f c) {
  // D = A(16x4 f32) * B(4x16 f32) + C(16x16 f32)
  return __builtin_amdgcn_wmma_f32_16x16x4_f32(false, a, false, b, (short)0, c,
                                               false, false);
}

// Async global->LDS copy, 16B per lane, GVS addressing (SGPR base + VGPR byte
// offset). LDS dest = low 32 bits of the generic pointer.
// Tracked by ASYNCcnt, no VGPR data round-trip.
__device__ __forceinline__ void async_copy16(void* lds_dst, const void* gbase,
                                             unsigned byte_off) {
  unsigned l = (unsigned)(uintptr_t)lds_dst;
  asm volatile("global_load_async_to_lds_b128 %0, %1, %2"
               :: "v"(l), "v"(byte_off), "s"(gbase)
               : "memory");
}

// Wait until at most N async copies are still in flight. Async loads complete
// in order, so N == copies-per-slab means "previous slab has fully landed".
template <int N>
__device__ __forceinline__ void async_wait() {
  asm volatile("s_wait_asynccnt %0" :: "n"(N) : "memory");
}

// ---------------- Kernel 1: g = x@Wg, u = x@Wu, act = silu(g)*u ----------------
__global__ __launch_bounds__(256) void moe_gate_up(
    const float* __restrict__ x,   // [E*T, H]
    const float* __restrict__ Wg,  // [E, H, F]
    const float* __restrict__ Wu,  // [E, H, F]
    float* __restrict__ act) {     // [E, T, F] (workspace)
  __shared__ float As[2][T_][KB + 4];   // row stride 20 floats
  __shared__ float Bgs[2][KB][NB + 4];  // row stride 132 floats
  __shared__ float Bus[2][KB][NB + 4];

  const int e   = blockIdx.y;
  const int f0  = blockIdx.x * NB;
  const int tid = threadIdx.x;
  const int w    = tid >> 5;
  const int lane = tid & 31;
  const int mt   = w & 3;              // M tile (16 rows each), 64 rows total
  const int nh   = w >> 2;             // 0..1 -> which 64-wide N half
  const int mrow = mt * 16 + (lane & 15);
  const int kb   = (lane >> 4) << 1;   // 0 or 2 (A/B frag K pair per half-wave)
  const int ncl  = lane & 15;

  const float* xe  = x  + (size_t)e * T_ * H_;
  const float* Wge = Wg + (size_t)e * H_ * F_ + f0;
  const float* Wue = Wu + (size_t)e * H_ * F_ + f0;

  // Staging geometry (per-thread, loop-invariant)
  const int ar  = tid >> 2;            // A row 0..63
  const int ac4 = (tid & 3) << 2;      // A col (float idx, 16B aligned)
  const int br  = tid >> 5;            // B row 0..7 (+8 on 2nd pass)
  const int bc4 = (tid & 31) << 2;     // B col

  // Issue one slab's async copies into buffer s (5 per thread: 1 A, 2 Bg, 2 Bu)
  auto prefetch = [&](int s, int k0) {
    async_copy16(&As[s][ar][ac4], xe, (unsigned)((ar * H_ + k0 + ac4) * 4));
#pragma unroll
    for (int i = 0; i < 2; ++i) {
      int r = br + i * 8;
      unsigned go = (unsigned)(((k0 + r) * F_ + bc4) * 4);
      async_copy16(&Bgs[s][r][bc4], Wge, go);
      async_copy16(&Bus[s][r][bc4], Wue, go);
    }
  };

  const v8f zero = {0.f, 0.f, 0.f, 0.f, 0.f, 0.f, 0.f, 0.f};
  v8f accg[4], accu[4];
#pragma unroll
  for (int j = 0; j < 4; ++j) { accg[j] = zero; accu[j] = zero; }

  prefetch(0, 0);
  int s = 0;
  for (int k0 = 0; k0 < H_; k0 += KB, s ^= 1) {
    if (k0 + KB < H_) {        // stream next slab while current one computes
      prefetch(s ^ 1, k0 + KB);
      async_wait<5>();         // current slab (older 5 copies) has landed
    } else {
      async_wait<0>();
    }
    __syncthreads();           // all waves' copies visible

#pragma unroll
    for (int k4 = 0; k4 < KB; k4 += 4) {
      v2f a;
      a.x = As[s][mrow][k4 + kb];
      a.y = As[s][mrow][k4 + kb + 1];
#pragma unroll
      for (int j = 0; j < 4; ++j) {
        int n = nh * 64 + j * 16 + ncl;
        v2f bg; bg.x = Bgs[s][k4 + kb][n]; bg.y = Bgs[s][k4 + kb + 1][n];
        accg[j] = wmma_f32(a, bg, accg[j]);
        v2f bu; bu.x = Bus[s][k4 + kb][n]; bu.y = Bus[s][k4 + kb + 1][n];
        accu[j] = wmma_f32(a, bu, accu[j]);
      }
    }
    __syncthreads();           // done reading buffer s before it is refilled
  }

  // Epilogue: act = silu(g) * u
  float* acte = act + (size_t)e * T_ * F_;
  const int thalf = (lane >> 4) << 3;  // 0 or 8
#pragma unroll
  for (int j = 0; j < 4; ++j) {
    int f = f0 + nh * 64 + j * 16 + ncl;
#pragma unroll
    for (int r = 0; r < 8; ++r) {
      int t = mt * 16 + thalf + r;
      float g = accg[j][r];
      float u = accu[j][r];
      float sg = g / (1.f + __expf(-g));  // silu(g)
      acte[(size_t)t * F_ + f] = sg * u;
    }
  }
}

// ---------------- Kernel 2: y = act @ Wd ----------------
__global__ __launch_bounds__(256) void moe_down(
    const float* __restrict__ act,  // [E, T, F]
    const float* __restrict__ Wd,   // [E, F, H]
    float* __restrict__ y) {        // [E*T, H]
  __shared__ float As[2][T_][KB + 4];
  __shared__ float Bs[2][KB][NB + 4];

  const int e   = blockIdx.y;
  const int h0  = blockIdx.x * NB;
  const int tid = threadIdx.x;
  const int w    = tid >> 5;
  const int lane = tid & 31;
  const int mt   = w & 3;
  const int nh   = w >> 2;
  const int mrow = mt * 16 + (lane & 15);
  const int kb   = (lane >> 4) << 1;
  const int ncl  = lane & 15;

  const float* ae  = act + (size_t)e * T_ * F_;
  const float* Wde = Wd + (size_t)e * F_ * H_ + h0;

  const int ar  = tid >> 2;
  const int ac4 = (tid & 3) << 2;
  const int br  = tid >> 5;
  const int bc4 = (tid & 31) << 2;

  // 3 async copies per thread per slab: 1 A, 2 B
  auto prefetch = [&](int s, int k0) {
    async_copy16(&As[s][ar][ac4], ae, (unsigned)((ar * F_ + k0 + ac4) * 4));
#pragma unroll
    for (int i = 0; i < 2; ++i) {
      int r = br + i * 8;
      async_copy16(&Bs[s][r][bc4], Wde, (unsigned)(((k0 + r) * H_ + bc4) * 4));
    }
  };

  const v8f zero = {0.f, 0.f, 0.f, 0.f, 0.f, 0.f, 0.f, 0.f};
  v8f acc[4];
#pragma unroll
  for (int j = 0; j < 4; ++j) acc[j] = zero;

  prefetch(0, 0);
  int s = 0;
  for (int k0 = 0; k0 < F_; k0 += KB, s ^= 1) {
    if (k0 + KB < F_) {
      prefetch(s ^ 1, k0 + KB);
      async_wait<3>();
    } else {
      async_wait<0>();
    }
    __syncthreads();

#pragma unroll
    for (int k4 = 0; k4 < KB; k4 += 4) {
      v2f a;
      a.x = As[s][mrow][k4 + kb];
      a.y = As[s][mrow][k4 + kb + 1];
#pragma unroll
      for (int j = 0; j < 4; ++j) {
        int n = nh * 64 + j * 16 + ncl;
        v2f b; b.x = Bs[s][k4 + kb][n]; b.y = Bs[s][k4 + kb + 1][n];
        acc[j] = wmma_f32(a, b, acc[j]);
      }
    }
    __syncthreads();
  }

  const int thalf = (lane >> 4) << 3;
#pragma unroll
  for (int j = 0; j < 4; ++j) {
    int h = h0 + nh * 64 + j * 16 + ncl;
#pragma unroll
    for (int r = 0; r < 8; ++r) {
      int t = mt * 16 + thalf + r;
      y[((size_t)e * T_ + t) * H_ + h] = acc[j][r];
    }
  }
}

extern "C" void kernel_launch(void* const* d_in, const int* in_sizes, int n_in,
                              void* d_out, int out_size, void* d_ws, size_t ws_size,
                              hipStream_t stream) {
  (void)in_sizes; (void)n_in; (void)out_size; (void)ws_size;
  const float* x  = (const float*)d_in[0];
  // d_in[1] = expert_tokens_count (int64, constant T per expert) -> unused
  const float* Wg = (const float*)d_in[2];
  const float* Wu = (const float*)d_in[3];
  const float* Wd = (const float*)d_in[4];
  float* act = (float*)d_ws;  // [E, T, F] f32 = 6.29 MB scratch

  moe_gate_up<<<dim3(F_ / NB, E_), 256, 0, stream>>>(x, Wg, Wu, act);
  moe_down<<<dim3(H_ / NB, E_), 256, 0, stream>>>(act, Wd, (float*)d_out);
}